// DynamicGraphLearner_9397388443889
// MI455X (gfx1250) — compile-verified
//
#include <hip/hip_runtime.h>
#include <cstddef>

// Shapes fixed by the reference: x [16, 256, 8, 128] -> B=16, M=2048, Fd=128
constexpr int BATCH = 16;
constexpr int M     = 2048;   // 256*8
constexpr int FD    = 128;
constexpr int NTILE = M / 16; // 128 column/row tiles of 16

typedef __attribute__((ext_vector_type(2))) float v2f;
typedef __attribute__((ext_vector_type(4))) float v4f;
typedef __attribute__((ext_vector_type(8))) float v8f;

// ---------------------------------------------------------------------------
// Kernel 1: L2-normalize each length-128 row (wave per row, wave32)
// ---------------------------------------------------------------------------
__global__ __launch_bounds__(256) void k_normalize(const float* __restrict__ x,
                                                   float* __restrict__ xn) {
    const int row  = blockIdx.x * (blockDim.x >> 5) + (threadIdx.x >> 5);
    const int lane = threadIdx.x & 31;
    const v4f v = *(const v4f*)(x + (size_t)row * FD + lane * 4);
    float ss = v.x * v.x + v.y * v.y + v.z * v.z + v.w * v.w;
#pragma unroll
    for (int m = 16; m >= 1; m >>= 1) ss += __shfl_xor(ss, m, 32);
    const float scale = 1.0f / fmaxf(sqrtf(ss), 1e-12f);
    v4f o;
    o.x = v.x * scale; o.y = v.y * scale; o.z = v.z * scale; o.w = v.w * scale;
    *(v4f*)(xn + (size_t)row * FD + lane * 4) = o;
}

// ---------------------------------------------------------------------------
// Kernel 2: zero the 256 MiB output with nontemporal 128-bit stores
// ---------------------------------------------------------------------------
__global__ __launch_bounds__(256) void k_zero(float* __restrict__ out, size_t n4) {
    size_t i = (size_t)blockIdx.x * blockDim.x + threadIdx.x;
    const size_t stride = (size_t)gridDim.x * blockDim.x;
    v4f z; z.x = 0.f; z.y = 0.f; z.z = 0.f; z.w = 0.f;
    for (; i < n4; i += stride)
        __builtin_nontemporal_store(z, (v4f*)out + i);
}

// ---------------------------------------------------------------------------
// Kernel 3: per-row top-2 of Xn·Xn^T via V_WMMA_F32_16X16X4_F32, then
//           symmetric scatter with atomics. One block per (batch, 16-row tile).
// ---------------------------------------------------------------------------
__global__ __launch_bounds__(256) void k_top2_scatter(const float* __restrict__ xn,
                                                      float* __restrict__ out) {
    const int b    = blockIdx.x >> 7;     // NTILE = 128 row tiles per batch
    const int rt   = blockIdx.x & 127;
    const int tid  = threadIdx.x;
    const int wave = tid >> 5;            // 8 waves
    const int lane = tid & 31;
    const int half = lane >> 4;           // 0: K even pair, 1: K odd pair (f32 A/B layout)
    const int sub  = lane & 15;           // M (for A) / N (for B) index within tile

    const float* xb      = xn + (size_t)b * M * FD;
    const int    rowbase = rt * 16;

    // Preload A tile: A[m=sub][k = kc*4 + 2*half + {0,1}]  (ISA 32-bit A layout)
    const float* Arow = xb + (size_t)(rowbase + sub) * FD + 2 * half;
    v2f a[FD / 4];
#pragma unroll
    for (int kc = 0; kc < FD / 4; ++kc) a[kc] = *(const v2f*)(Arow + kc * 4);

    // Per-lane running top-2 for the 8 local rows this lane holds in C/D
    float v1[8], v2r[8];
    int   i1[8], i2[8];
#pragma unroll
    for (int g = 0; g < 8; ++g) { v1[g] = -3.0f; v2r[g] = -3.0f; i1[g] = 0; i2[g] = 0; }

    // Each wave sweeps column tiles ct = wave, wave+8, ...  (uniform per wave)
    for (int ct = wave; ct < NTILE; ct += 8) {
        const float* Brow = xb + (size_t)(ct * 16 + sub) * FD + 2 * half;
        v8f acc = {0.f, 0.f, 0.f, 0.f, 0.f, 0.f, 0.f, 0.f};
#pragma unroll
        for (int kc = 0; kc < FD / 4; ++kc) {
            v2f bv = *(const v2f*)(Brow + kc * 4);
            acc = __builtin_amdgcn_wmma_f32_16x16x4_f32(
                false, a[kc], false, bv, (short)0, acc, false, false);
        }
        // Reference zeroes the diagonal BEFORE top-k: diag candidate value = 0.0
        if (ct == rt) {
#pragma unroll
            for (int g = 0; g < 8; ++g)
                if ((g + 8 * half) == sub) acc[g] = 0.0f;
        }
        const int c = ct * 16 + sub;   // column this lane holds in C/D
#pragma unroll
        for (int g = 0; g < 8; ++g) {
            const float val = acc[g];
            if (val > v1[g])       { v2r[g] = v1[g]; i2[g] = i1[g]; v1[g] = val; i1[g] = c; }
            else if (val > v2r[g]) { v2r[g] = val;   i2[g] = c; }
        }
    }

    // Butterfly merge of sorted top-2 pairs across the 16 lanes of each half
#pragma unroll
    for (int m = 1; m <= 8; m <<= 1) {
#pragma unroll
        for (int g = 0; g < 8; ++g) {
            const float c1 = __shfl_xor(v1[g],  m, 32);
            const float c2 = __shfl_xor(v2r[g], m, 32);
            const int   d1 = __shfl_xor(i1[g],  m, 32);
            const int   d2 = __shfl_xor(i2[g],  m, 32);
            float n1, n2; int j1, j2;
            if ((v1[g] > c1) || (v1[g] == c1 && i1[g] <= d1)) {
                n1 = v1[g]; j1 = i1[g];
                if ((c1 > v2r[g]) || (c1 == v2r[g] && d1 < i2[g])) { n2 = c1; j2 = d1; }
                else                                              { n2 = v2r[g]; j2 = i2[g]; }
            } else {
                n1 = c1; j1 = d1;
                if ((v1[g] > c2) || (v1[g] == c2 && i1[g] < d2))  { n2 = v1[g]; j2 = i1[g]; }
                else                                              { n2 = c2; j2 = d2; }
            }
            v1[g] = n1; i1[g] = j1; v2r[g] = n2; i2[g] = j2;
        }
    }

    __shared__ float sV1[8][16], sV2[8][16];
    __shared__ int   sI1[8][16], sI2[8][16];
    if (sub == 0) {
#pragma unroll
        for (int g = 0; g < 8; ++g) {
            const int r = g + 8 * half;   // local row in tile (C layout: M = g + 8*half)
            sV1[wave][r] = v1[g];  sV2[wave][r] = v2r[g];
            sI1[wave][r] = i1[g];  sI2[wave][r] = i2[g];
        }
    }
    __syncthreads();

    // One thread per row: merge the 8 wave candidates, scatter symmetrically
    if (tid < 16) {
        const int r = tid;
        float b1 = -4.0f, b2 = -4.0f; int x1 = 0, x2 = 0;
#pragma unroll
        for (int w = 0; w < 8; ++w) {
            float cv = sV1[w][r]; int ci = sI1[w][r];
            if (cv > b1 || (cv == b1 && ci < x1))      { b2 = b1; x2 = x1; b1 = cv; x1 = ci; }
            else if (cv > b2 || (cv == b2 && ci < x2)) { b2 = cv; x2 = ci; }
            cv = sV2[w][r]; ci = sI2[w][r];
            if (cv > b1 || (cv == b1 && ci < x1))      { b2 = b1; x2 = x1; b1 = cv; x1 = ci; }
            else if (cv > b2 || (cv == b2 && ci < x2)) { b2 = cv; x2 = ci; }
        }
        const int grow = rowbase + r;
        float* ob = out + (size_t)b * M * M;
        // out = (S + S^T)/2 over zeroed buffer -> 0.5*v at (r,i) and (i,r)
        atomicAdd(&ob[(size_t)grow * M + x1], 0.5f * b1);
        atomicAdd(&ob[(size_t)x1 * M + grow], 0.5f * b1);
        atomicAdd(&ob[(size_t)grow * M + x2], 0.5f * b2);
        atomicAdd(&ob[(size_t)x2 * M + grow], 0.5f * b2);
    }
}

// ---------------------------------------------------------------------------
extern "C" void kernel_launch(void* const* d_in, const int* in_sizes, int n_in,
                              void* d_out, int out_size, void* d_ws, size_t ws_size,
                              hipStream_t stream) {
    (void)in_sizes; (void)n_in; (void)out_size; (void)ws_size;
    const float* x   = (const float*)d_in[0];
    float*       out = (float*)d_out;
    float*       xnw = (float*)d_ws;                 // 16 MiB normalized copy

    // K1: 32768 rows, 8 rows (waves) per 256-thread block
    k_normalize<<<(BATCH * M) / 8, 256, 0, stream>>>(x, xnw);

    // K2: zero 16*2048*2048 floats = 16,777,216 float4s
    const size_t n4 = (size_t)BATCH * M * M / 4;
    k_zero<<<4096, 256, 0, stream>>>(out, n4);

    // K3: one block per (batch, 16-row tile)
    k_top2_scatter<<<BATCH * NTILE, 256, 0, stream>>>(xnw, out);
}